// AttentionReadoutAtom_4561255268925
// MI455X (gfx1250) — compile-verified
//
#include <hip/hip_runtime.h>

#define IN_DIM 128

typedef float v2f __attribute__((ext_vector_type(2)));
typedef float v8f __attribute__((ext_vector_type(8)));

// Order-preserving mapping float -> uint so unsigned atomicMax gives float max.
__device__ __forceinline__ unsigned f32_to_key(float f) {
  unsigned b = __float_as_uint(f);
  return (b & 0x80000000u) ? ~b : (b | 0x80000000u);
}
__device__ __forceinline__ float key_to_f32(unsigned k) {
  return __uint_as_float((k & 0x80000000u) ? (k & 0x7FFFFFFFu) : ~k);
}

// ---------------------------------------------------------------------------
// Pass 1: scores[i] = dot(x[i,:], w) + b via V_WMMA_F32_16X16X4_F32.
// One wave32 handles 16 rows. A-tile (16x4 f32): lanes 0-15 hold K=k0,k0+1;
// lanes 16-31 hold K=k0+2,k0+3 (per CDNA5 ISA layout). B-tile replicates the
// same w-chunk in every N column, so every column of D equals the score.
// Lanes with N==0 (lanes 0 and 16) own rows 0-7 / 8-15 across acc[0..7].
// ---------------------------------------------------------------------------
__global__ __launch_bounds__(256) void k_scores(
    const float* __restrict__ x, const float* __restrict__ w,
    const float* __restrict__ bptr, float* __restrict__ scores,
    unsigned* __restrict__ maxKey, int nTiles) {
  const int wavesPerBlk = blockDim.x >> 5;
  const int tile = blockIdx.x * wavesPerBlk + ((int)threadIdx.x >> 5);
  if (tile >= nTiles) return;                 // wave-uniform: EXEC all-1s for WMMA
  const int lane = (int)threadIdx.x & 31;
  const int h = lane >> 4;                    // 0: lanes 0-15, 1: lanes 16-31
  const int m = lane & 15;                    // A-matrix row within tile
  const long rowBase = (long)tile * 16;

  const float* rp = x + (rowBase + m) * IN_DIM + 2 * h;   // 8B aligned
  const float* wp = w + 2 * h;                            // 8B aligned

  v8f acc = {};
#pragma unroll
  for (int k0 = 0; k0 < IN_DIM; k0 += 4) {
    v2f a = *(const v2f*)(rp + k0);           // A: x[row, k0+2h .. k0+2h+1]
    v2f b = *(const v2f*)(wp + k0);           // B: w[k0+2h .. k0+2h+1], all columns
    acc = __builtin_amdgcn_wmma_f32_16x16x4_f32(
        /*neg_a=*/false, a, /*neg_b=*/false, b,
        /*c_mod=*/(short)0, acc, /*reuse_a=*/false, /*reuse_b=*/false);
  }

  if (m == 0) {                               // lanes 0 (rows 0-7) and 16 (rows 8-15)
    const float bias = bptr[0];
    float lmax = -3.402823466e38f;
#pragma unroll
    for (int r = 0; r < 8; ++r) {
      float s = acc[r] + bias;
      scores[rowBase + 8 * h + r] = s;
      lmax = fmaxf(lmax, s);
    }
    atomicMax(maxKey, f32_to_key(lmax));
  }
}

// ---------------------------------------------------------------------------
// Pass 2: sum = sum_i exp(scores[i] - max). Wave32 shuffle reduce + atomicAdd.
// ---------------------------------------------------------------------------
__global__ __launch_bounds__(256) void k_sumexp(
    const float* __restrict__ scores, const unsigned* __restrict__ maxKey,
    float* __restrict__ sumOut, int n) {
  const float maxv = key_to_f32(*maxKey);
  float local = 0.f;
  const long stride = (long)gridDim.x * blockDim.x;
  for (long i = (long)blockIdx.x * blockDim.x + threadIdx.x; i < n; i += stride)
    local += __expf(scores[i] - maxv);
#pragma unroll
  for (int off = 16; off > 0; off >>= 1)
    local += __shfl_xor(local, off, 32);
  if (((int)threadIdx.x & 31) == 0) atomicAdd(sumOut, local);
}

// ---------------------------------------------------------------------------
// Pass 3: out[label[i], :] += x[i, :] * exp(s_i - max) / sum.
// One wave per row: lane loads float4 -> the whole 512B row is one coalesced
// wave transaction; 4 global_atomic_add_f32 per lane land in L2 (out fits).
// ---------------------------------------------------------------------------
__global__ __launch_bounds__(256) void k_scatter(
    const float* __restrict__ x, const int* __restrict__ labels,
    const float* __restrict__ scores, const unsigned* __restrict__ maxKey,
    const float* __restrict__ sumPtr, float* __restrict__ out, int n) {
  const int wavesPerBlk = blockDim.x >> 5;
  const long row = (long)blockIdx.x * wavesPerBlk + ((int)threadIdx.x >> 5);
  if (row >= n) return;
  const int lane = (int)threadIdx.x & 31;

  const float maxv = key_to_f32(*maxKey);
  const float invSum = 1.f / sumPtr[0];
  const float a = __expf(scores[row] - maxv) * invSum;

  const float4 xv = ((const float4*)(x + row * IN_DIM))[lane];
  const int seg = labels[row];
  float* op = out + (long)seg * IN_DIM + lane * 4;
  atomicAdd(op + 0, xv.x * a);
  atomicAdd(op + 1, xv.y * a);
  atomicAdd(op + 2, xv.z * a);
  atomicAdd(op + 3, xv.w * a);
}

// ---------------------------------------------------------------------------
extern "C" void kernel_launch(void* const* d_in, const int* in_sizes, int n_in,
                              void* d_out, int out_size, void* d_ws, size_t ws_size,
                              hipStream_t stream) {
  const float* x      = (const float*)d_in[0];
  const int*   labels = (const int*)d_in[1];
  const float* w      = (const float*)d_in[2];
  const float* b      = (const float*)d_in[3];
  float* out = (float*)d_out;
  const int n = in_sizes[0] / IN_DIM;         // 500000 rows

  // Workspace: [0]=max key (u32), [1]=sum (f32), scores[] at +16 bytes.
  unsigned* maxKey = (unsigned*)d_ws;
  float*    sum    = (float*)d_ws + 1;
  float*    scores = (float*)d_ws + 4;

  hipMemsetAsync(d_out, 0, (size_t)out_size * sizeof(float), stream);
  hipMemsetAsync(d_ws, 0, 16, stream);        // maxKey=0 (== -inf key), sum=0

  const int nTiles  = (n + 15) / 16;          // N is an exact multiple of 16
  const int blocks1 = (nTiles + 7) / 8;       // 8 waves per 256-thread block
  k_scores<<<blocks1, 256, 0, stream>>>(x, w, b, scores, maxKey, nTiles);

  k_sumexp<<<1024, 256, 0, stream>>>(scores, maxKey, sum, n);

  const int blocks3 = (n + 7) / 8;            // 8 rows (waves) per block
  k_scatter<<<blocks3, 256, 0, stream>>>(x, labels, scores, maxKey, sum, out, n);
}